// P2RNRebuildModule_57251914056191
// MI455X (gfx1250) — compile-verified
//
#include <hip/hip_runtime.h>
#include <math.h>

#define NB 128
#define HH 256
#define WW 256
#define HWP (HH * WW)
#define KPT 68
#define OSCALE 6.0f

typedef __attribute__((ext_vector_type(2))) float v2f;
typedef __attribute__((ext_vector_type(8))) float v8f;

#if defined(__gfx1250__) && __has_builtin(__builtin_amdgcn_wmma_f32_16x16x4_f32)
#define USE_WMMA 1
#else
#define USE_WMMA 0
#endif

// ---------------------------------------------------------------------------
// Kernel A: one block per batch. Gather 68 keypoints, estimate scaled rigid
// transform. R = (U@Vh)^T == transpose of the orthogonal polar factor of the
// 3x3 cross-covariance; computed via det-scaled Newton polar iteration in f64.
// Writes per-batch [Rs(3x3 row-major) | t(3)] = 12 floats to RT.
// ---------------------------------------------------------------------------
__global__ void __launch_bounds__(128) tform_kernel(
    const float* __restrict__ Offset, const float* __restrict__ Posmap,
    const float* __restrict__ meanp, const int* __restrict__ uv,
    float* __restrict__ RT)
{
  __shared__ float s_src[KPT][3];
  __shared__ float s_dst[KPT][3];
  __shared__ int s_i64;

  const int b = blockIdx.x;

  // Detect whether uv buffer is int32 (136 ints) or little-endian int64
  // (272 ints, every odd slot == 0 since values < 256).
  if (threadIdx.x == 0) {
    int allz = 1;
    for (int i = 1; i < 2 * KPT; i += 2) {
      if (uv[i] != 0) { allz = 0; break; }
    }
    s_i64 = allz;
  }
  __syncthreads();

  const int k = threadIdx.x;
  if (k < KPT) {
    int u0, u1;
    if (s_i64) { u0 = uv[4 * k + 0]; u1 = uv[4 * k + 2]; }
    else       { u0 = uv[2 * k + 0]; u1 = uv[2 * k + 1]; }
    const int pix = u0 * WW + u1;
    const size_t ob = (size_t)b * 3 * HWP;
    #pragma unroll
    for (int c = 0; c < 3; ++c) {
      s_src[k][c] = Offset[ob + (size_t)c * HWP + pix] * OSCALE +
                    meanp[(size_t)c * HWP + pix];
      s_dst[k][c] = Posmap[ob + (size_t)c * HWP + pix];
    }
  }
  __syncthreads();
  if (threadIdx.x != 0) return;

  // ---- serial f64 solve (trivial cost: 128 blocks x ~3k flops) ----
  double msrc[3] = {0, 0, 0}, mdst[3] = {0, 0, 0};
  double sd1 = 0.0, sd2 = 0.0;
  double c33[3], d33[3];
  for (int c = 0; c < 3; ++c) { c33[c] = s_src[33][c]; d33[c] = s_dst[33][c]; }
  for (int kk = 0; kk < KPT; ++kk) {
    double s0 = s_src[kk][0], s1 = s_src[kk][1], s2 = s_src[kk][2];
    double t0 = s_dst[kk][0], t1 = s_dst[kk][1], t2 = s_dst[kk][2];
    double e0 = s0 - c33[0], e1 = s1 - c33[1], e2 = s2 - c33[2];
    double f0 = t0 - d33[0], f1 = t1 - d33[1], f2 = t2 - d33[2];
    sd1 += sqrt(e0 * e0 + e1 * e1 + e2 * e2);
    sd2 += sqrt(f0 * f0 + f1 * f1 + f2 * f2);
    msrc[0] += s0; msrc[1] += s1; msrc[2] += s2;
    mdst[0] += t0; mdst[1] += t1; mdst[2] += t2;
  }
  for (int c = 0; c < 3; ++c) { msrc[c] /= (double)KPT; mdst[c] /= (double)KPT; }

  // centered cross-covariance C[i][j] = sum_k (src_i - msrc_i)(dst_j - mdst_j)
  double C[3][3] = {{0,0,0},{0,0,0},{0,0,0}};
  for (int kk = 0; kk < KPT; ++kk) {
    double a0 = s_src[kk][0] - msrc[0];
    double a1 = s_src[kk][1] - msrc[1];
    double a2 = s_src[kk][2] - msrc[2];
    double b0 = s_dst[kk][0] - mdst[0];
    double b1 = s_dst[kk][1] - mdst[1];
    double b2 = s_dst[kk][2] - mdst[2];
    C[0][0] += a0 * b0; C[0][1] += a0 * b1; C[0][2] += a0 * b2;
    C[1][0] += a1 * b0; C[1][1] += a1 * b1; C[1][2] += a1 * b2;
    C[2][0] += a2 * b0; C[2][1] += a2 * b1; C[2][2] += a2 * b2;
  }
  const double s = sd2 / sd1;  // Hm = s*C; polar factor invariant to s>0

  // det-scaled Newton polar iteration: X <- 0.5*(g*X + (1/g)*X^{-T})
  double X[3][3];
  for (int i = 0; i < 3; ++i)
    for (int j = 0; j < 3; ++j) X[i][j] = C[i][j];
  for (int it = 0; it < 20; ++it) {
    double cof[3][3];
    cof[0][0] =  X[1][1]*X[2][2] - X[1][2]*X[2][1];
    cof[0][1] = -(X[1][0]*X[2][2] - X[1][2]*X[2][0]);
    cof[0][2] =  X[1][0]*X[2][1] - X[1][1]*X[2][0];
    cof[1][0] = -(X[0][1]*X[2][2] - X[0][2]*X[2][1]);
    cof[1][1] =  X[0][0]*X[2][2] - X[0][2]*X[2][0];
    cof[1][2] = -(X[0][0]*X[2][1] - X[0][1]*X[2][0]);
    cof[2][0] =  X[0][1]*X[1][2] - X[0][2]*X[1][1];
    cof[2][1] = -(X[0][0]*X[1][2] - X[0][2]*X[1][0]);
    cof[2][2] =  X[0][0]*X[1][1] - X[0][1]*X[1][0];
    double det = X[0][0]*cof[0][0] + X[0][1]*cof[0][1] + X[0][2]*cof[0][2];
    double ad = fabs(det);
    if (ad < 1e-300) ad = 1e-300;
    double g  = 1.0 / cbrt(ad);     // makes |det(g*X)| = 1
    double gi = 1.0 / (g * det);    // (1/g) * X^{-T} = gi * cof
    for (int i = 0; i < 3; ++i)
      for (int j = 0; j < 3; ++j)
        X[i][j] = 0.5 * (g * X[i][j] + gi * cof[i][j]);
  }
  // Q = X = polar factor (== U@Vh).  R[j][i] = Q[i][j];  Rs = s*R.
  float* rt = RT + b * 12;
  for (int j = 0; j < 3; ++j) {
    double tj = mdst[j] - s * (msrc[0]*X[0][j] + msrc[1]*X[1][j] + msrc[2]*X[2][j]);
    rt[9 + j] = (float)tj;
    for (int i = 0; i < 3; ++i) rt[j * 3 + i] = (float)(s * X[i][j]);
  }
}

// ---------------------------------------------------------------------------
// Kernel B: memory-bound affine apply. One V_WMMA_F32_16X16X4_F32 applies the
// 3x4 affine [Rs|t] (in A, rows 0-2) to 16 pixels (in B, rows 0-2 = planes,
// row 3 = ones). 4 tiles (64 px) per wave per pass, 8 waves/block.
// A layout (16x4 f32): lanes 0-15 hold {K0,K1}, lanes 16-31 hold {K2,K3}.
// B layout (4x16 f32): lanes 0-15 = K rows 0/1, lanes 16-31 = K rows 2/3.
// D rows 0-2 -> d[0..2] of lanes 0-15 (contiguous 16-px stores per plane).
// ---------------------------------------------------------------------------
__global__ void __launch_bounds__(256) apply_kernel(
    const float* __restrict__ Offset, const float* __restrict__ meanp,
    const float* __restrict__ RT, float* __restrict__ out)
{
  const int b = blockIdx.y;
  const float* __restrict__ rt = RT + b * 12;
  const float* __restrict__ o0 = Offset + (size_t)b * 3 * HWP;
  float* __restrict__ q0 = out + (size_t)b * 3 * HWP;

#if USE_WMMA
  const int lane = threadIdx.x & 31;
  const int wave = threadIdx.x >> 5;
  const int half = lane >> 4;
  const int l    = lane & 15;
  const int base = blockIdx.x * 512 + wave * 64;

  v2f a; a.x = 0.0f; a.y = 0.0f;
  if (l < 3) {
    if (half) { a.x = rt[l * 3 + 2]; a.y = rt[9 + l]; }      // K2=Rs[l][2], K3=t[l]
    else      { a.x = rt[l * 3 + 0]; a.y = rt[l * 3 + 1]; }  // K0,K1
  }

  #pragma unroll
  for (int t = 0; t < 4; ++t) {
    const int p = base + t * 16 + l;
    v2f bv;
    if (half) {
      bv.x = __builtin_nontemporal_load(o0 + 2 * HWP + p) * OSCALE + meanp[2 * HWP + p];
      bv.y = 1.0f;                                            // ones row (K=3)
    } else {
      bv.x = __builtin_nontemporal_load(o0 + p) * OSCALE + meanp[p];
      bv.y = __builtin_nontemporal_load(o0 + HWP + p) * OSCALE + meanp[HWP + p];
    }
    v8f c = {};
    v8f d = __builtin_amdgcn_wmma_f32_16x16x4_f32(
        /*neg_a=*/false, a, /*neg_b=*/false, bv,
        /*c_mod=*/(short)0, c, /*reuse_a=*/false, /*reuse_b=*/false);
    if (!half) {
      __builtin_nontemporal_store(d[0], q0 + p);
      __builtin_nontemporal_store(d[1], q0 + HWP + p);
      __builtin_nontemporal_store(d[2], q0 + 2 * HWP + p);
    }
  }
#else
  // Exact-math fallback: 2 pixels/thread, 9 FMAs each.
  const float r00 = rt[0], r01 = rt[1], r02 = rt[2];
  const float r10 = rt[3], r11 = rt[4], r12 = rt[5];
  const float r20 = rt[6], r21 = rt[7], r22 = rt[8];
  const float t0 = rt[9], t1 = rt[10], t2 = rt[11];
  const int p0 = blockIdx.x * 512 + (int)threadIdx.x * 2;
  #pragma unroll
  for (int u = 0; u < 2; ++u) {
    const int p = p0 + u;
    float x0 = o0[p] * OSCALE + meanp[p];
    float x1 = o0[HWP + p] * OSCALE + meanp[HWP + p];
    float x2 = o0[2 * HWP + p] * OSCALE + meanp[2 * HWP + p];
    q0[p]           = r00 * x0 + r01 * x1 + r02 * x2 + t0;
    q0[HWP + p]     = r10 * x0 + r11 * x1 + r12 * x2 + t1;
    q0[2 * HWP + p] = r20 * x0 + r21 * x1 + r22 * x2 + t2;
  }
#endif
}

extern "C" void kernel_launch(void* const* d_in, const int* in_sizes, int n_in,
                              void* d_out, int out_size, void* d_ws, size_t ws_size,
                              hipStream_t stream) {
  const float* Offset = (const float*)d_in[0];
  const float* Posmap = (const float*)d_in[1];
  const float* meanp  = (const float*)d_in[2];
  const int*   uv     = (const int*)d_in[3];
  float* RT  = (float*)d_ws;          // NB * 12 floats = 6 KB
  float* out = (float*)d_out;

  tform_kernel<<<dim3(NB), dim3(128), 0, stream>>>(Offset, Posmap, meanp, uv, RT);
  apply_kernel<<<dim3(HWP / 512, NB), dim3(256), 0, stream>>>(Offset, meanp, RT, out);
}